// RGCNEncoder_89962384982025
// MI455X (gfx1250) — compile-verified
//
#include <hip/hip_runtime.h>
#include <stddef.h>

// Problem constants (match reference)
#define N_NODES 50000
#define N_REL   16
#define N_BASES 8
#define DIM     128
#define N_EDGES 625000
#define N_W     17              // 16 relations + self-loop weight
#define LN_EPS  1e-5f

typedef float v2f __attribute__((ext_vector_type(2)));
typedef float v8f __attribute__((ext_vector_type(8)));

// ---------------------------------------------------------------------------
// deg[r*N + row] += 1 per edge (stored as float; later inverted in place)
// ---------------------------------------------------------------------------
__global__ __launch_bounds__(256) void rgcn_deg_kernel(
    const int* __restrict__ edge_index, const int* __restrict__ edge_type,
    float* __restrict__ deg) {
  int e = blockIdx.x * blockDim.x + threadIdx.x;
  if (e >= N_EDGES) return;
  int r = edge_index[e];          // row
  int t = edge_type[e];
  atomicAdd(deg + (size_t)t * N_NODES + r, 1.0f);
}

// deg -> winv = 1/max(deg,1)
__global__ __launch_bounds__(256) void rgcn_winv_kernel(float* __restrict__ d) {
  int i = blockIdx.x * blockDim.x + threadIdx.x;
  if (i >= N_REL * N_NODES) return;
  d[i] = 1.0f / fmaxf(d[i], 1.0f);
}

// ---------------------------------------------------------------------------
// Wt[r][o][i] = sum_b coef[r,b]*bases[b,i,o]   (transposed so GEMM B-frags are
// contiguous float2 along k).  r==16 slot holds selfw (transposed).
// ---------------------------------------------------------------------------
__global__ __launch_bounds__(256) void rgcn_wprep_kernel(
    const float* __restrict__ bases, const float* __restrict__ coef,
    const float* __restrict__ selfw, float* __restrict__ Wt) {
  int idx = blockIdx.x * blockDim.x + threadIdx.x;
  if (idx >= N_W * DIM * DIM) return;
  int r  = idx >> 14;             // /(128*128)
  int oi = idx & (DIM * DIM - 1);
  int o  = oi >> 7;               // output col (n)
  int i  = oi & 127;              // input row  (k)
  float s;
  if (r < N_REL) {
    s = 0.0f;
#pragma unroll
    for (int b = 0; b < N_BASES; ++b)
      s += coef[r * N_BASES + b] * bases[b * DIM * DIM + i * DIM + o];
  } else {
    s = selfw[i * DIM + o];
  }
  Wt[idx] = s;                    // layout [r][o][i]
}

// ---------------------------------------------------------------------------
// Scatter: T[et, row, :] += x[col, :].  One wave32 per edge, float4 per lane.
// x (25.6MB) is L2-resident; atomics stream into T.
// ---------------------------------------------------------------------------
__global__ __launch_bounds__(256) void rgcn_scatter_kernel(
    const int* __restrict__ edge_index, const int* __restrict__ edge_type,
    const float* __restrict__ x, float* __restrict__ T) {
  int gid  = blockIdx.x * blockDim.x + threadIdx.x;
  int e    = gid >> 5;
  int lane = gid & 31;
  if (e >= N_EDGES) return;
  int t = edge_type[e];
  int r = edge_index[e];              // row (destination segment)
  int c = edge_index[N_EDGES + e];    // col (gathered source)
  const float4 v = *(const float4*)(x + (size_t)c * DIM + lane * 4);
  float* dst = T + ((size_t)t * N_NODES + r) * DIM + lane * 4;
  atomicAdd(dst + 0, v.x);
  atomicAdd(dst + 1, v.y);
  atomicAdd(dst + 2, v.z);
  atomicAdd(dst + 3, v.w);
}

// ---------------------------------------------------------------------------
// Fused: out[n,:] = ReLU(LN( bias + x@selfw + sum_r (winv[r,n]*T[r,n,:])@W[r] ))
// Workgroup: 256 threads = 8 waves, computes a 16-row x 128-col tile.
// Wave w owns col tile [16w,16w+16) via V_WMMA_F32_16X16X4_F32, K accumulated
// in 32 steps of 4.  A tile staged in LDS (132-float stride, conflict-free
// ds_load_b64), scaled by winv on the way in.  B read as contiguous float2
// from the transposed, L2-resident Wt.
// ---------------------------------------------------------------------------
__global__ __launch_bounds__(256) void rgcn_gemm_ln_kernel(
    const float* __restrict__ x, const float* __restrict__ T,
    const float* __restrict__ Wt, const float* __restrict__ winv,
    const float* __restrict__ bias, const float* __restrict__ gam,
    const float* __restrict__ bet, float* __restrict__ out) {
  __shared__ float As[16 * 132];      // A tile (reused as C tile for LN)

  const int tid  = threadIdx.x;
  const int wave = tid >> 5;
  const int lane = tid & 31;
  const int row0 = blockIdx.x * 16;
  const int n0   = wave * 16;         // wave's 16-col slice
  const int m    = lane & 15;         // WMMA m (A) / n (B) index
  const int kh   = (lane >> 4) << 1;  // lanes 16-31 handle k+2,k+3

  // cooperative A loader mapping: thread -> (row 0..15, 8 consecutive cols)
  const int arow = tid >> 4;
  const int acol = (tid & 15) * 8;

  v8f acc = {};                       // 16x16 f32 accumulator (8 VGPRs)

  for (int r = 0; r < N_W; ++r) {
    // ---- stage A tile (winv-scaled) into LDS ----
    const float* asrc;
    float scale;
    if (r < N_REL) {
      asrc  = T + ((size_t)r * N_NODES + row0 + arow) * DIM + acol;
      scale = winv[r * N_NODES + row0 + arow];
    } else {
      asrc  = x + (size_t)(row0 + arow) * DIM + acol;
      scale = 1.0f;
    }
    float4 a0 = *(const float4*)(asrc);
    float4 a1 = *(const float4*)(asrc + 4);
    float* ad = As + arow * 132 + acol;
    ad[0] = a0.x * scale; ad[1] = a0.y * scale;
    ad[2] = a0.z * scale; ad[3] = a0.w * scale;
    ad[4] = a1.x * scale; ad[5] = a1.y * scale;
    ad[6] = a1.z * scale; ad[7] = a1.w * scale;
    if (r + 1 < N_REL)                // stream-prefetch next relation's rows
      __builtin_prefetch(T + ((size_t)(r + 1) * N_NODES + row0 + arow) * DIM + acol, 0, 1);
    __syncthreads();

    // ---- 32 x WMMA f32 16x16x4 over K=128 ----
    const float* wr = Wt + (size_t)r * (DIM * DIM) + (n0 + m) * DIM;  // B col (n0+m)
#pragma unroll
    for (int k = 0; k < DIM; k += 4) {
      v2f av = *(const v2f*)(As + m * 132 + k + kh);  // A[m][k+kh], A[m][k+kh+1]
      v2f bv = *(const v2f*)(wr + k + kh);            // B[k+kh][n],  B[k+kh+1][n]
      acc = __builtin_amdgcn_wmma_f32_16x16x4_f32(
          false, av, false, bv, (short)0, acc, false, false);
    }
    __syncthreads();                  // before next r overwrites As
  }

  // ---- epilogue: +bias, spill C tile to LDS, LayerNorm + ReLU ----
  const float bv = bias[n0 + m];      // column is fixed per lane across acc[v]
#pragma unroll
  for (int v = 0; v < 8; ++v) {
    int mm = v + ((lane >> 4) << 3);  // C layout: VGPR v -> M = v (+8 for hi half)
    As[mm * 132 + n0 + m] = acc[v] + bv;
  }
  __syncthreads();

  const float4 g4 = *(const float4*)(gam + lane * 4);
  const float4 b4 = *(const float4*)(bet + lane * 4);
#pragma unroll
  for (int rr2 = 0; rr2 < 2; ++rr2) { // each wave normalizes 2 rows
    int rr = wave * 2 + rr2;
    float4 val = *(const float4*)(As + rr * 132 + lane * 4);
    float s = val.x + val.y + val.z + val.w;
    float q = val.x * val.x + val.y * val.y + val.z * val.z + val.w * val.w;
#pragma unroll
    for (int off = 16; off > 0; off >>= 1) {
      s += __shfl_xor(s, off, 32);
      q += __shfl_xor(q, off, 32);
    }
    float mean = s * (1.0f / 128.0f);
    float var  = q * (1.0f / 128.0f) - mean * mean;
    float rstd = rsqrtf(var + LN_EPS);
    float4 o;
    o.x = fmaxf(0.0f, (val.x - mean) * rstd * g4.x + b4.x);
    o.y = fmaxf(0.0f, (val.y - mean) * rstd * g4.y + b4.y);
    o.z = fmaxf(0.0f, (val.z - mean) * rstd * g4.z + b4.z);
    o.w = fmaxf(0.0f, (val.w - mean) * rstd * g4.w + b4.w);
    *(float4*)(out + (size_t)(row0 + rr) * DIM + lane * 4) = o;
  }
}

// ---------------------------------------------------------------------------
extern "C" void kernel_launch(void* const* d_in, const int* in_sizes, int n_in,
                              void* d_out, int out_size, void* d_ws, size_t ws_size,
                              hipStream_t stream) {
  (void)in_sizes; (void)n_in; (void)out_size; (void)ws_size;
  const int*   edge_index = (const int*)d_in[0];
  const int*   edge_type  = (const int*)d_in[1];
  const float* emb        = (const float*)d_in[2];
  const float* bases1     = (const float*)d_in[3];
  const float* coef1      = (const float*)d_in[4];
  const float* selfw1     = (const float*)d_in[5];
  const float* bias1      = (const float*)d_in[6];
  const float* g1         = (const float*)d_in[7];
  const float* b1         = (const float*)d_in[8];
  const float* bases2     = (const float*)d_in[9];
  const float* coef2      = (const float*)d_in[10];
  const float* selfw2     = (const float*)d_in[11];
  const float* bias2      = (const float*)d_in[12];
  const float* g2         = (const float*)d_in[13];
  const float* b2         = (const float*)d_in[14];
  float* out = (float*)d_out;

  // workspace layout (~440 MB total)
  float* ws = (float*)d_ws;
  const size_t T_elems = (size_t)N_REL * N_NODES * DIM;   // 102.4M floats
  float* T    = ws;
  float* winv = T + T_elems;                              // 800k floats
  float* W    = winv + (size_t)N_REL * N_NODES;           // 278528 floats
  float* x1   = W + (size_t)N_W * DIM * DIM;              // 6.4M floats

  // edge degrees -> winv (shared by both layers)
  hipMemsetAsync(winv, 0, (size_t)N_REL * N_NODES * sizeof(float), stream);
  rgcn_deg_kernel<<<(N_EDGES + 255) / 256, 256, 0, stream>>>(edge_index, edge_type, winv);
  rgcn_winv_kernel<<<(N_REL * N_NODES) / 256, 256, 0, stream>>>(winv);

  // ---- layer 1 ----
  hipMemsetAsync(T, 0, T_elems * sizeof(float), stream);
  rgcn_wprep_kernel<<<(N_W * DIM * DIM) / 256, 256, 0, stream>>>(bases1, coef1, selfw1, W);
  rgcn_scatter_kernel<<<(N_EDGES * 32) / 256, 256, 0, stream>>>(edge_index, edge_type, emb, T);
  rgcn_gemm_ln_kernel<<<N_NODES / 16, 256, 0, stream>>>(emb, T, W, winv, bias1, g1, b1, x1);

  // ---- layer 2 ----
  hipMemsetAsync(T, 0, T_elems * sizeof(float), stream);
  rgcn_wprep_kernel<<<(N_W * DIM * DIM) / 256, 256, 0, stream>>>(bases2, coef2, selfw2, W);
  rgcn_scatter_kernel<<<(N_EDGES * 32) / 256, 256, 0, stream>>>(edge_index, edge_type, x1, T);
  rgcn_gemm_ln_kernel<<<N_NODES / 16, 256, 0, stream>>>(x1, T, W, winv, bias2, g2, b2, out);
}